// ModelParallelCrossEntropy_3899830304796
// MI455X (gfx1250) — compile-verified
//
#include <hip/hip_runtime.h>

// Problem constants from the reference: logits [1024, 128000] fp32, labels int.
#define CE_B   1024
#define CE_V   128000
#define CE_V4  (CE_V / 4)     // 32000 float4 per row
#define CE_BLK 512            // 16 wave32 waves per workgroup

typedef float v4f __attribute__((ext_vector_type(4)));

// Online-softmax combine: (m1,s1) ⊕ (m2,s2) -> (max, rescaled sum)
__device__ __forceinline__ void ce_combine(float& m1, float& s1, float m2, float s2) {
    float mn = fmaxf(m1, m2);
    s1 = s1 * __expf(m1 - mn) + s2 * __expf(m2 - mn);
    m1 = mn;
}

// Process one float4 against the running (m, s) state.
__device__ __forceinline__ void ce_accum(const v4f& x, float& m, float& s) {
    float m4 = fmaxf(fmaxf(x.x, x.y), fmaxf(x.z, x.w));
    if (m4 > m) {            // rare: rescale running sum to new max
        s *= __expf(m - m4); // exp(-inf)=0 handles the first iteration
        m = m4;
    }
    // steady state: exactly one v_exp_f32 per element
    s += __expf(x.x - m) + __expf(x.y - m) + __expf(x.z - m) + __expf(x.w - m);
}

// One workgroup per row: single streaming pass with online max/sum.
// NT-hinted b128 loads, WGP-scope prefetch ahead, software-pipelined loop,
// deterministic LDS tree reduction.
__global__ __launch_bounds__(CE_BLK)
void ce_row_kernel(const float* __restrict__ logits,
                   const int*   __restrict__ labels,
                   float*       __restrict__ row_loss) {
    __shared__ float sm[CE_BLK];
    __shared__ float ss[CE_BLK];

    const int row = blockIdx.x;
    const int tid = threadIdx.x;
    const v4f* __restrict__ rowp =
        reinterpret_cast<const v4f*>(logits + (size_t)row * CE_V);

    float m = -INFINITY;
    float s = 0.0f;

    // Software-pipelined streaming loop: issue next NT b128 load before
    // consuming the current one, so VALU (exp/max) overlaps HBM latency
    // within a single wave as well as across the 16 waves of the block.
    int i = tid;                                   // always < CE_V4 (32000)
    v4f cur = __builtin_nontemporal_load(rowp + i);
    while (i + CE_BLK < CE_V4) {
        // gfx1250 prefetch (global_prefetch_b8), WGP scope: fill all levels.
        int pf = i + CE_BLK * 8;
        if (pf < CE_V4) __builtin_prefetch((const void*)(rowp + pf), 0, 3);

        v4f nxt = __builtin_nontemporal_load(rowp + i + CE_BLK);
        ce_accum(cur, m, s);
        cur = nxt;
        i += CE_BLK;
    }
    ce_accum(cur, m, s);

    sm[tid] = m;
    ss[tid] = s;
    __syncthreads();

    // Deterministic LDS tree combine of (m, s) pairs.
    for (int off = CE_BLK / 2; off > 0; off >>= 1) {
        if (tid < off) {
            float m1 = sm[tid], s1 = ss[tid];
            ce_combine(m1, s1, sm[tid + off], ss[tid + off]);
            sm[tid] = m1;
            ss[tid] = s1;
        }
        __syncthreads();
    }

    if (tid == 0) {
        const int lbl = labels[row];
        const float xl = logits[(size_t)row * CE_V + (size_t)lbl];
        // loss_row = log(sum exp(x - m)) + m - x_label
        row_loss[row] = logf(ss[0]) + sm[0] - xl;
    }
}

// Single-block deterministic mean over the 1024 per-row losses.
__global__ __launch_bounds__(256)
void ce_final_kernel(const float* __restrict__ row_loss,
                     float*       __restrict__ out) {
    __shared__ float sh[256];
    const int tid = threadIdx.x;

    float s = 0.0f;
    for (int i = tid; i < CE_B; i += 256)   // fixed order -> deterministic
        s += row_loss[i];
    sh[tid] = s;
    __syncthreads();

    for (int off = 128; off > 0; off >>= 1) {
        if (tid < off) sh[tid] += sh[tid + off];
        __syncthreads();
    }
    if (tid == 0) out[0] = sh[0] * (1.0f / (float)CE_B);
}

extern "C" void kernel_launch(void* const* d_in, const int* in_sizes, int n_in,
                              void* d_out, int out_size, void* d_ws, size_t ws_size,
                              hipStream_t stream) {
    const float* logits = (const float*)d_in[0];  // [1024, 128000] fp32
    const int*   labels = (const int*)d_in[1];    // [1024] int
    float*       out    = (float*)d_out;          // scalar loss
    float*       ws     = (float*)d_ws;           // >= 1024 floats of scratch

    ce_row_kernel<<<CE_B, CE_BLK, 0, stream>>>(logits, labels, ws);
    ce_final_kernel<<<1, 256, 0, stream>>>(ws, out);
}